// CriterionPC_79491254714952
// MI455X (gfx1250) — compile-verified
//
#include <hip/hip_runtime.h>
#include <math.h>

// ---------------------------------------------------------------------------
// CriterionPC loss on MI455X (gfx1250, wave32).
// Pass 1: per-block partial sums of normalized feats per (batch,class,channel).
// Pass 1b: finalize means + their norms.
// Pass 2: per-pixel cosines via V_WMMA_F32_16X16X4_F32 (feat[16x16] x mean[16x2]),
//         exp / squared-diff, block partial loss.
// Pass 3: final mean.
// Bandwidth-bound: ~530 MiB traffic => ~23 us at 23.3 TB/s. Native v_rcp_f32 /
// v_sqrt_f32 keep per-byte VALU overhead off the critical path (ROCm __fdividef
// is NOT fast without -ffast-math; use amdgcn builtins directly).
// ---------------------------------------------------------------------------

typedef float v2f __attribute__((ext_vector_type(2)));
typedef float v8f __attribute__((ext_vector_type(8)));

#define NBATCH 32
#define NCH    16
#define NPIX   65536            // 256*256 pixels per image
#define NCLS   2

// phase-1 geometry: 8 blocks/batch * 256 thr * 32 px = 65536 px
#define P1_BLOCKS_PER_BATCH 8
#define P1_THREADS 256
#define P1_PIX_PER_THREAD 32
#define P1_VALS 66              // 4*16 channel sums + 2 counts

// phase-2 geometry: 32 blocks/batch; 8 waves * 16 tiles * 16 px = 2048 px/block
#define P2_BLOCKS_PER_BATCH 32
#define P2_THREADS 256
#define P2_WAVES 8
#define P2_TILES 16

// workspace layout (floats)
#define WS_PART1   0
#define WS_PART1_N (NBATCH * P1_BLOCKS_PER_BATCH * P1_VALS)
#define WS_MEANS   (WS_PART1 + WS_PART1_N)          // [b][64]: S0,S1,T0,T1 x 16ch
#define WS_MEANS_N (NBATCH * 64)
#define WS_NRMS    (WS_MEANS + WS_MEANS_N)          // [b][4]: |mS0|,|mS1|,|mT0|,|mT1|
#define WS_NRMS_N  (NBATCH * 4)
#define WS_LOSS    (WS_NRMS + WS_NRMS_N)            // [1024] block partial losses
#define WS_LOSS_N  (NBATCH * P2_BLOCKS_PER_BATCH)

__device__ __forceinline__ float waveRedAdd(float v) {
#pragma unroll
  for (int m = 16; m > 0; m >>= 1) v += __shfl_xor(v, m, 32);
  return v;
}

// fast a/b via native v_rcp_f32 (~1 ulp; operands here are O(1))
__device__ __forceinline__ float fastDiv(float a, float b) {
  return a * __builtin_amdgcn_rcpf(b);
}

// ---------------------------------------------------------------------------
// Phase 1: per-pixel L2-normalize over channels, accumulate per-class sums.
// Deterministic: register accumulators -> wave shuffle -> LDS -> block partial.
// ---------------------------------------------------------------------------
__global__ __launch_bounds__(P1_THREADS)
void k_phase1(const float* __restrict__ S, const float* __restrict__ T,
              const int* __restrict__ tgt, float* __restrict__ part) {
  const int b   = blockIdx.y;
  const int tid = threadIdx.x;

  float aS0[NCH] = {}, aS1[NCH] = {}, aT0[NCH] = {}, aT1[NCH] = {};
  float cnt0 = 0.f, cnt1 = 0.f;

  const size_t base = (size_t)b * NCH * NPIX;
  const int*   tb   = tgt + (size_t)b * NPIX;

  for (int i = 0; i < P1_PIX_PER_THREAD; ++i) {
    const int pix = blockIdx.x * (P1_THREADS * P1_PIX_PER_THREAD) + i * P1_THREADS + tid;
    const float w1 = (tb[pix] != 0) ? 1.f : 0.f;
    const float w0 = 1.f - w1;
    cnt0 += w0; cnt1 += w1;

    float x[NCH]; float ss = 0.f;
#pragma unroll
    for (int c = 0; c < NCH; ++c) { x[c] = S[base + (size_t)c * NPIX + pix]; ss = fmaf(x[c], x[c], ss); }
    // F.normalize: x / max(||x||, 1e-12)
    float inv = __builtin_amdgcn_rcpf(fmaxf(__builtin_amdgcn_sqrtf(ss), 1e-12f));
#pragma unroll
    for (int c = 0; c < NCH; ++c) { float f = x[c] * inv; aS0[c] = fmaf(w0, f, aS0[c]); aS1[c] = fmaf(w1, f, aS1[c]); }

    ss = 0.f;
#pragma unroll
    for (int c = 0; c < NCH; ++c) { x[c] = T[base + (size_t)c * NPIX + pix]; ss = fmaf(x[c], x[c], ss); }
    inv = __builtin_amdgcn_rcpf(fmaxf(__builtin_amdgcn_sqrtf(ss), 1e-12f));
#pragma unroll
    for (int c = 0; c < NCH; ++c) { float f = x[c] * inv; aT0[c] = fmaf(w0, f, aT0[c]); aT1[c] = fmaf(w1, f, aT1[c]); }
  }

  float v[P1_VALS];
#pragma unroll
  for (int c = 0; c < NCH; ++c) { v[c] = aS0[c]; v[16 + c] = aS1[c]; v[32 + c] = aT0[c]; v[48 + c] = aT1[c]; }
  v[64] = cnt0; v[65] = cnt1;

  __shared__ float sred[P1_THREADS / 32][P1_VALS];
  const int w = tid >> 5, l = tid & 31;
#pragma unroll
  for (int j = 0; j < P1_VALS; ++j) {
    float r = waveRedAdd(v[j]);
    if (l == 0) sred[w][j] = r;
  }
  __syncthreads();
  if (tid < P1_VALS) {
    float s = 0.f;
#pragma unroll
    for (int ww = 0; ww < P1_THREADS / 32; ++ww) s += sred[ww][tid];
    part[(size_t)(b * P1_BLOCKS_PER_BATCH + blockIdx.x) * P1_VALS + tid] = s;
  }
}

// ---------------------------------------------------------------------------
// Phase 1b: reduce block partials -> means [b][64] and mean norms [b][4].
// ---------------------------------------------------------------------------
__global__ __launch_bounds__(96)
void k_means(const float* __restrict__ part, float* __restrict__ means,
             float* __restrict__ nrms) {
  const int b = blockIdx.x, t = threadIdx.x;
  __shared__ float scnt[2];
  __shared__ float smean[64];

  float sum = 0.f;
  if (t < P1_VALS) {
#pragma unroll
    for (int blk = 0; blk < P1_BLOCKS_PER_BATCH; ++blk)
      sum += part[(size_t)(b * P1_BLOCKS_PER_BATCH + blk) * P1_VALS + t];
  }
  if (t == 64) scnt[0] = sum;
  if (t == 65) scnt[1] = sum;
  __syncthreads();
  if (t < 64) {
    const int cls = (t >> 4) & 1;                 // layout: S0,S1,T0,T1
    const float m = sum / (scnt[cls] + 1e-6f);    // MEAN_EPS (not perf-critical)
    smean[t] = m;
    means[b * 64 + t] = m;
  }
  __syncthreads();
  if (t < 4) {
    float ss = 0.f;
#pragma unroll
    for (int k = 0; k < 16; ++k) { float m = smean[t * 16 + k]; ss = fmaf(m, m, ss); }
    nrms[b * 4 + t] = sqrtf(ss);
  }
}

// ---------------------------------------------------------------------------
// Phase 2: per-pixel cosines via chained V_WMMA_F32_16X16X4_F32.
//   A (16x4 f32): lanes 0-15 hold M=lane, K=0,1 (2 VGPRs); lanes 16-31 K=2,3.
//   B (4x16 f32): lanes 0-15 hold K=0,1 for col N=lane; lanes 16-31 K=2,3.
//   D: lane n / n+16 hold col n rows 0-7 / 8-15 across 8 VGPRs.
// Columns 0/1 of B carry mean[class0/1]; all other columns zero.
// ---------------------------------------------------------------------------
__global__ __launch_bounds__(P2_THREADS)
void k_phase2(const float* __restrict__ S, const float* __restrict__ T,
              const int* __restrict__ tgt,
              const float* __restrict__ means, const float* __restrict__ nrms,
              float* __restrict__ lossPart) {
  const int b     = blockIdx.y;
  const int tid   = threadIdx.x;
  const int w     = tid >> 5, l = tid & 31;
  const int ncol  = l & 15;        // pixel-in-tile for A rows / column for B
  const int khalf = l >> 4;        // which K pair this lane provides

  __shared__ float smean[64];
  __shared__ float snrm[4];
  __shared__ float sdot[P2_WAVES][2][2][16];   // [wave][S/T][class][pixel]
  __shared__ float sredw[P2_WAVES];

  if (tid < 64) smean[tid] = means[b * 64 + tid];
  if (tid < 4)  snrm[tid]  = nrms[b * 4 + tid];
  __syncthreads();

  // B operands: constant per block (chunk kc covers channels 4kc..4kc+3)
  v2f bS[4], bT[4];
#pragma unroll
  for (int kc = 0; kc < 4; ++kc) {
    const int cb = 4 * kc + 2 * khalf;
    float s0 = 0.f, s1 = 0.f, t0 = 0.f, t1 = 0.f;
    if (ncol < NCLS) {
      s0 = smean[ncol * 16 + cb];      s1 = smean[ncol * 16 + cb + 1];
      t0 = smean[32 + ncol * 16 + cb]; t1 = smean[32 + ncol * 16 + cb + 1];
    }
    bS[kc].x = s0; bS[kc].y = s1;
    bT[kc].x = t0; bT[kc].y = t1;
  }

  const size_t base     = (size_t)b * NCH * NPIX;
  const int    blockPix = blockIdx.x * (P2_WAVES * P2_TILES * 16);
  float lossAcc = 0.f;

  for (int it = 0; it < P2_TILES; ++it) {
    const int p0  = blockPix + w * (P2_TILES * 16) + it * 16;
    const int pix = p0 + ncol;

    // prefetch next tile's cachelines (global_prefetch_b8)
    __builtin_prefetch(&S[base + (size_t)(2 * khalf) * NPIX + pix + 16], 0, 3);
    __builtin_prefetch(&T[base + (size_t)(2 * khalf) * NPIX + pix + 16], 0, 3);

    v2f aS[4], aT[4];
    float ssS = 0.f, ssT = 0.f;
#pragma unroll
    for (int kc = 0; kc < 4; ++kc) {
      const int ch = 4 * kc + 2 * khalf;
      float x0 = S[base + (size_t)ch * NPIX + pix];
      float x1 = S[base + (size_t)(ch + 1) * NPIX + pix];
      ssS = fmaf(x0, x0, fmaf(x1, x1, ssS));
      aS[kc].x = x0; aS[kc].y = x1;
      float y0 = T[base + (size_t)ch * NPIX + pix];
      float y1 = T[base + (size_t)(ch + 1) * NPIX + pix];
      ssT = fmaf(y0, y0, fmaf(y1, y1, ssT));
      aT[kc].x = y0; aT[kc].y = y1;
    }
    // combine the two half-norms held by lane pairs (l, l^16)
    ssS += __shfl_xor(ssS, 16, 32);
    ssT += __shfl_xor(ssT, 16, 32);
    const float nS = __builtin_amdgcn_sqrtf(ssS), nT = __builtin_amdgcn_sqrtf(ssT);
    const float invS = __builtin_amdgcn_rcpf(fmaxf(nS, 1e-12f));
    const float invT = __builtin_amdgcn_rcpf(fmaxf(nT, 1e-12f));
    const float naS = nS * invS, naT = nT * invT;   // ||feat|| (≈1)
#pragma unroll
    for (int kc = 0; kc < 4; ++kc) { aS[kc] *= invS; aT[kc] *= invT; }

    v8f accS = {0.f, 0.f, 0.f, 0.f, 0.f, 0.f, 0.f, 0.f};
    v8f accT = {0.f, 0.f, 0.f, 0.f, 0.f, 0.f, 0.f, 0.f};
#pragma unroll
    for (int kc = 0; kc < 4; ++kc) {
      accS = __builtin_amdgcn_wmma_f32_16x16x4_f32(false, aS[kc], false, bS[kc],
                                                   (short)0, accS, false, false);
      accT = __builtin_amdgcn_wmma_f32_16x16x4_f32(false, aT[kc], false, bT[kc],
                                                   (short)0, accT, false, false);
    }

    // scatter D columns 0/1 (the two class dots) to pixel-owning lanes via LDS
    if (ncol < NCLS) {
#pragma unroll
      for (int r = 0; r < 8; ++r) {
        sdot[w][0][ncol][khalf * 8 + r] = accS[r];
        sdot[w][1][ncol][khalf * 8 + r] = accT[r];
      }
    }
    __syncthreads();

    if (l < 16) {
      const int m   = l;
      const int cls = tgt[(size_t)b * NPIX + p0 + m];
      const float dSc = sdot[w][0][cls][m],     dSf = sdot[w][0][1 - cls][m];
      const float dTc = sdot[w][1][cls][m],     dTf = sdot[w][1][1 - cls][m];
      const float cSc = fastDiv(dSc, fmaxf(naS * snrm[cls],     1e-8f));
      const float cSf = fastDiv(dSf, fmaxf(naS * snrm[1 - cls], 1e-8f));
      const float cTc = fastDiv(dTc, fmaxf(naT * snrm[2 + cls], 1e-8f));
      const float cTf = fastDiv(dTf, fmaxf(naT * snrm[3 - cls], 1e-8f));
      const float d = __expf(cSc - cSf) - __expf(cTc - cTf);
      lossAcc = fmaf(d, d, lossAcc);
    }
    __syncthreads();
  }

  float r = waveRedAdd(lossAcc);
  if (l == 0) sredw[w] = r;
  __syncthreads();
  if (tid == 0) {
    float s = 0.f;
#pragma unroll
    for (int ww = 0; ww < P2_WAVES; ++ww) s += sredw[ww];
    lossPart[b * P2_BLOCKS_PER_BATCH + blockIdx.x] = s;
  }
}

// ---------------------------------------------------------------------------
// Phase 3: final mean over all block partials.
// ---------------------------------------------------------------------------
__global__ __launch_bounds__(256)
void k_final(const float* __restrict__ lossPart, float* __restrict__ out) {
  float s = 0.f;
  for (int i = threadIdx.x; i < WS_LOSS_N; i += 256) s += lossPart[i];
  s = waveRedAdd(s);
  __shared__ float sredw[8];
  const int w = threadIdx.x >> 5, l = threadIdx.x & 31;
  if (l == 0) sredw[w] = s;
  __syncthreads();
  if (threadIdx.x == 0) {
    float t = 0.f;
#pragma unroll
    for (int ww = 0; ww < 8; ++ww) t += sredw[ww];
    out[0] = t / (float)((size_t)NBATCH * NPIX);
  }
}

extern "C" void kernel_launch(void* const* d_in, const int* in_sizes, int n_in,
                              void* d_out, int out_size, void* d_ws, size_t ws_size,
                              hipStream_t stream) {
  (void)in_sizes; (void)n_in; (void)out_size; (void)ws_size;
  const float* S   = (const float*)d_in[0];
  const float* T   = (const float*)d_in[1];
  const int*   tgt = (const int*)d_in[2];

  float* ws       = (float*)d_ws;
  float* part     = ws + WS_PART1;
  float* means    = ws + WS_MEANS;
  float* nrms     = ws + WS_NRMS;
  float* lossPart = ws + WS_LOSS;

  k_phase1<<<dim3(P1_BLOCKS_PER_BATCH, NBATCH), P1_THREADS, 0, stream>>>(S, T, tgt, part);
  k_means <<<dim3(NBATCH), 96, 0, stream>>>(part, means, nrms);
  k_phase2<<<dim3(P2_BLOCKS_PER_BATCH, NBATCH), P2_THREADS, 0, stream>>>(S, T, tgt, means, nrms, lossPart);
  k_final <<<1, 256, 0, stream>>>(lossPart, (float*)d_out);
}